// Hydra_44160853737521
// MI455X (gfx1250) — compile-verified
//
#include <hip/hip_runtime.h>

// ---------------------------------------------------------------------------
// Hydra-style multi-dilation grouped conv + argmax/argmin histogram, CDNA5.
// One wave32 per (dilation, diff, group, batch) combo. The 9-tap x 8-kernel
// depthwise conv is computed as chained V_WMMA_F32_16X16X4_F32 ops:
//   A (16x4 fp32) = kernel weights (rows 0-7 = kernels, K = tap chunk)
//   B (4x16 fp32) = sel patches for 16 output positions (zero-padded bounds)
//   D (16x16)     = Z[kernel, position] accumulated over 3 tap chunks
// fp32 WMMA keeps bit-accuracy for the tie-sensitive argmax/argmin.
// B-operand bounds handling is branchless (clamped LDS index + cndmask) so
// EXEC stays all-ones through the hot loop; position loop unrolled x2 so the
// VALU reduction of one strip overlaps the WMMA triple of the next.
// ---------------------------------------------------------------------------

typedef float v2f __attribute__((ext_vector_type(2)));
typedef float v8f __attribute__((ext_vector_type(8)));

#define BB    64
#define CC    12
#define LL    2048
#define KERN9 9
#define KK    8     // kernels per group
#define GG    32    // groups
#define NPER  6     // channels summed per group
#define NDIL  8

__global__ __launch_bounds__(32)
void hydra_wmma_kernel(const float* __restrict__ X,   // [B, C, L]
                       const float* __restrict__ W,   // [NDIL, 2, KK*GG, 1, 9]
                       const int*   __restrict__ I,   // [NDIL, 2, GG, NPER]
                       float*       __restrict__ out) // [B, NDIL*2*2*GG*KK]
{
    __shared__ float sel[LL];   // 8 KB of the 320 KB/WGP LDS

    const int cid  = blockIdx.x;          // 15 bits: di|diff|g|b
    const int b    = cid & 63;
    const int g    = (cid >> 6) & 31;
    const int diff = (cid >> 11) & 1;
    const int di   = (cid >> 12) & 7;
    const int d    = 1 << di;             // dilation
    const int Lp   = diff ? (LL - 1) : LL;

    const int lane = threadIdx.x;         // wave32: 0..31
    const int m    = lane & 15;           // WMMA row / column index within half
    const int hi   = lane >> 4;           // 0: K pair (0,1); 1: K pair (2,3)

    // ---- channel indices for this (di, diff, g) -----------------------------
    const int* Ip = I + (((di * 2 + diff) * GG) + g) * NPER;
    int ch[NPER];
#pragma unroll
    for (int n = 0; n < NPER; ++n) ch[n] = Ip[n];

    // ---- build sel[] in LDS (channel-subset sum, optional first difference) -
    const float* Xb = X + (size_t)b * CC * LL;
    for (int l = lane; l < LL; l += 32) {
        float v = 0.f;
        if (l < Lp) {
            if (diff == 0) {
#pragma unroll
                for (int n = 0; n < NPER; ++n) v += Xb[ch[n] * LL + l];
            } else {
#pragma unroll
                for (int n = 0; n < NPER; ++n)
                    v += Xb[ch[n] * LL + l + 1] - Xb[ch[n] * LL + l];
            }
        }
        sel[l] = v;   // positions >= Lp stay 0 (zero padding)
    }
    __syncthreads();

    // ---- load weights into A-matrix register layout -------------------------
    // 16x4 fp32 A: lanes 0-15 hold (K=0,K=1) in (v0,v1); lanes 16-31 hold (K=2,K=3).
    // Rows 8-15 (unused kernel slots) and taps >= 9 are zero.
    const float* Wp = W + ((size_t)(di * 2 + diff) * (KK * GG) + (size_t)g * KK) * KERN9;
    v2f A[3];
#pragma unroll
    for (int c = 0; c < 3; ++c) {
        const int t0 = 4 * c + 2 * hi;
        float ax = 0.f, ay = 0.f;
        if (m < KK) {
            if (t0     < KERN9) ax = Wp[m * KERN9 + t0];
            if (t0 + 1 < KERN9) ay = Wp[m * KERN9 + t0 + 1];
        }
        A[c].x = ax;
        A[c].y = ay;
    }

    // ---- per-lane histograms ------------------------------------------------
    float accmax[KK], accmin[KK];
#pragma unroll
    for (int j = 0; j < KK; ++j) { accmax[j] = 0.f; accmin[j] = 0.f; }

    const bool laneHalf0 = (lane < 16);   // only lanes 0-15 carry valid D rows 0-7

    // ---- main loop: 128 strips of 16 output positions -----------------------
#pragma unroll 2
    for (int base = 0; base < LL; base += 16) {
        v8f c8 = {};
#pragma unroll
        for (int c = 0; c < 3; ++c) {
            // B (4x16 fp32): v0 holds rows K=0 (lanes 0-15) / K=2 (lanes 16-31),
            //                v1 holds rows K=1 / K=3; column = lane & 15.
            // Branchless zero padding: clamp index, always load, then cndmask.
            const int t0 = 4 * c + 2 * hi;
            const int i0 = base + m + (t0     - 4) * d;
            const int i1 = base + m + (t0 + 1 - 4) * d;
            const int c0 = min(max(i0, 0), LL - 1);
            const int c1 = min(max(i1, 0), LL - 1);
            const float v0 = sel[c0];
            const float v1 = sel[c1];
            v2f Bv;
            Bv.x = (i0 >= 0 && i0 < Lp) ? v0 : 0.f;  // zero-pad = conv padding
            Bv.y = (i1 >= 0 && i1 < Lp) ? v1 : 0.f;
            c8 = __builtin_amdgcn_wmma_f32_16x16x4_f32(
                     /*neg_a=*/false, A[c], /*neg_b=*/false, Bv,
                     /*c_mod=*/(short)0, c8, /*reuse_a=*/false, /*reuse_b=*/false);
        }

        // lanes 0-15: c8[k] = Z[kernel k, position base+m]  (D rows 0-7)
        float mx = c8[0]; int amax = 0;
        float mn = c8[0]; int amin = 0;
#pragma unroll
        for (int k = 1; k < KK; ++k) {
            const float zv = c8[k];
            if (zv > mx) { mx = zv; amax = k; }   // first-max, matches jnp.argmax
            if (zv < mn) { mn = zv; amin = k; }   // first-min, matches jnp.argmin
        }
        const bool valid = laneHalf0 && ((base + m) < Lp);
        const float mxv = valid ? mx : 0.f;       // fold validity into operands
        const float one = valid ? 1.f : 0.f;
#pragma unroll
        for (int j = 0; j < KK; ++j) {
            accmax[j] += (amax == j) ? mxv : 0.f;
            accmin[j] += (amin == j) ? one : 0.f;
        }
    }

    // ---- cross-lane position reduction (lanes 16-31 contribute zeros) -------
#pragma unroll
    for (int j = 0; j < KK; ++j) {
        float am = accmax[j], an = accmin[j];
#pragma unroll
        for (int off = 16; off >= 1; off >>= 1) {
            am += __shfl_xor(am, off, 32);
            an += __shfl_xor(an, off, 32);
        }
        accmax[j] = am;
        accmin[j] = an;
    }

    // ---- write: out[b, ((di*2+diff)*2 + which)*GG + g, k] -------------------
    if (lane == 0) {
        const size_t chanStride = (size_t)NDIL * 2 * 2 * GG * KK;  // 8192
        size_t ob = (size_t)b * chanStride
                  + ((size_t)((di * 2 + diff) * 2 + 0) * GG + g) * KK;
#pragma unroll
        for (int j = 0; j < KK; ++j) out[ob + j] = accmax[j];
        ob = (size_t)b * chanStride
           + ((size_t)((di * 2 + diff) * 2 + 1) * GG + g) * KK;
#pragma unroll
        for (int j = 0; j < KK; ++j) out[ob + j] = accmin[j];
    }
}

extern "C" void kernel_launch(void* const* d_in, const int* in_sizes, int n_in,
                              void* d_out, int out_size, void* d_ws, size_t ws_size,
                              hipStream_t stream) {
    const float* X = (const float*)d_in[0];   // [64, 12, 2048] fp32
    const float* W = (const float*)d_in[1];   // [8, 2, 256, 1, 9] fp32
    const int*   I = (const int*)d_in[2];     // [8, 2, 32, 6] int32
    float* out = (float*)d_out;               // [64, 8192] fp32

    const int numCombos = NDIL * 2 * GG * BB; // 32768 single-wave workgroups
    hydra_wmma_kernel<<<dim3(numCombos), dim3(32), 0, stream>>>(X, W, I, out);
}